// Scattering_Classifier_17755394801982
// MI455X (gfx1250) — compile-verified
//
#include <hip/hip_runtime.h>
#include <math.h>

// ---------------- problem constants ----------------
#define BB      32
#define TT      16000         // = 2^7 * 5^3
#define NN1     120
#define TT2     2000          // = 2^4 * 5^3
#define NPAIRS  553
#define NFEAT   673           // 120 + 553
#define FBAND   96            // phi_hat sigma = 6.4 bins -> 15 sigma cutoff (exact to fp32)
#define BN1     (BB*NN1)      // 3840
#define BNP     (BB*NPAIRS)   // 17696

// padded MLP dims (multiples of 16 so the WMMA loop is branch-free)
#define KP1     688           // 673 -> 688 = 16*43
#define NP1     304           // 300 -> 304 = 16*19
#define NP2     96            // 90  -> 96  = 16*6

#define PI_F     3.14159265358979323846f
#define TWO_PI_F 6.28318530717958647692f

typedef __attribute__((ext_vector_type(2))) float v2f;
typedef __attribute__((ext_vector_type(8))) float v8f;

// ---------------- complex helpers ----------------
__device__ __forceinline__ float2 expif(float a) { float s, c; __sincosf(a, &s, &c); return make_float2(c, s); }
__device__ __forceinline__ float2 cmul(float2 a, float2 b) { return make_float2(a.x*b.x - a.y*b.y, a.x*b.y + a.y*b.x); }
__device__ __forceinline__ float2 cadd(float2 a, float2 b) { return make_float2(a.x + b.x, a.y + b.y); }
__device__ __forceinline__ float2 csub(float2 a, float2 b) { return make_float2(a.x - b.x, a.y - b.y); }

__device__ __forceinline__ float wave_all_sum(float v) {
#pragma unroll
  for (int off = 16; off > 0; off >>= 1) v += __shfl_xor(v, off, 32);
  return v;
}

// ---------------- real -> complex ----------------
__global__ void k_r2c(const float* __restrict__ in, float2* __restrict__ out, long long n) {
  long long i = (long long)blockIdx.x * blockDim.x + threadIdx.x;
  if (i < n) out[i] = make_float2(in[i], 0.f);
}

// ---------------- batched mixed-radix Stockham pass (radix 2 or 5) ----------------
__global__ void k_stockham(const float2* __restrict__ src, float2* __restrict__ dst,
                           int N, int Ns, int radix, int nbatch, float sign) {
  int perB = N / radix;
  long long t = (long long)blockIdx.x * blockDim.x + threadIdx.x;
  if (t >= (long long)nbatch * perB) return;
  int b = (int)(t / perB);
  int j = (int)(t % perB);
  const float2* x = src + (size_t)b * N;
  float2*       y = dst + (size_t)b * N;
  int jmod = j % Ns;
  float ang = sign * TWO_PI_F * (float)jmod / (float)(Ns * radix);
  if (radix == 2) {
    float2 v0 = x[j];
    float2 v1 = cmul(x[j + perB], expif(ang));
    int id = (j / Ns) * (Ns * 2) + jmod;
    y[id]      = cadd(v0, v1);
    y[id + Ns] = csub(v0, v1);
  } else { // radix 5
    float2 v[5];
#pragma unroll
    for (int r = 0; r < 5; ++r) v[r] = cmul(x[j + r * perB], expif(ang * (float)r));
    float2 w1 = expif(sign * TWO_PI_F / 5.f);
    float2 w2 = cmul(w1, w1);
    float2 w3 = cmul(w2, w1);
    float2 w4 = cmul(w3, w1);
    float2 w[5] = { make_float2(1.f, 0.f), w1, w2, w3, w4 };
    int id = (j / Ns) * (Ns * 5) + jmod;
#pragma unroll
    for (int k = 0; k < 5; ++k) {
      float2 acc = v[0];
#pragma unroll
      for (int r = 1; r < 5; ++r) acc = cadd(acc, cmul(v[r], w[(k * r) % 5]));
      y[id + k * Ns] = acc;
    }
  }
}

// ---------------- stage-1 modulation: Xf[b] * psi1_hat[n] ----------------
__global__ void k_mod1(const float2* __restrict__ Xf, const float* __restrict__ psi1,
                       float2* __restrict__ out) {
  long long i = (long long)blockIdx.x * blockDim.x + threadIdx.x;
  if (i >= (long long)BN1 * TT) return;
  int f  = (int)(i % TT);
  int bn = (int)(i / TT);
  int b = bn / NN1, n = bn % NN1;
  float2 xv = Xf[(size_t)b * TT + f];
  float  p  = psi1[(size_t)n * TT + f];
  out[i] = make_float2(xv.x * p, xv.y * p);
}

// ---------------- |ifft| (scale 1/T) + subsample-by-8 ----------------
__global__ void k_abs1(const float2* __restrict__ in, float* __restrict__ U1,
                       float* __restrict__ U1S) {
  long long i = (long long)blockIdx.x * blockDim.x + threadIdx.x;
  if (i >= (long long)BN1 * TT) return;
  float2 v = in[i];
  float a = sqrtf(v.x * v.x + v.y * v.y) * (1.f / (float)TT);
  U1[i] = a;
  int t = (int)(i % TT);
  if ((t & 7) == 0) {
    long long bn = i / TT;
    U1S[bn * TT2 + (t >> 3)] = a;
  }
}

// ---------------- band-limited forward DFT (one wave per (bn,f), f<96) ----------------
__global__ void k_banddft(const float* __restrict__ u, float2* __restrict__ C, int len, int nb) {
  long long gt = (long long)blockIdx.x * blockDim.x + threadIdx.x;
  int gw   = (int)(gt >> 5);
  int lane = threadIdx.x & 31;
  if (gw >= nb * FBAND) return;
  int bn = gw / FBAND, f = gw % FBAND;
  const float* row = u + (size_t)bn * len;
  float w = -TWO_PI_F * (float)f / (float)len;
  float sr = 0.f, si = 0.f;
  for (int t = lane; t < len; t += 32) {
    float s, c; __sincosf(w * (float)t, &s, &c);
    float uv = row[t];
    sr = fmaf(uv, c, sr);
    si = fmaf(uv, s, si);
  }
  sr = wave_all_sum(sr);
  si = wave_all_sum(si);
  if (lane == 0) C[(size_t)gw] = make_float2(sr, si);
}

// ---------------- lowpass eval at 16 taps + log + mean ----------------
__global__ void k_finalize(const float2* __restrict__ C, const float* __restrict__ phi,
                           float invLen, int nb, int perB, int featOff,
                           float* __restrict__ Fout) {
  long long gt = (long long)blockIdx.x * blockDim.x + threadIdx.x;
  int gw   = (int)(gt >> 5);
  int lane = threadIdx.x & 31;
  if (gw >= nb) return;
  float2 cf[3]; float fv[3];
#pragma unroll
  for (int jj = 0; jj < 3; ++jj) {
    int f = lane + 32 * jj;
    float2 c = C[(size_t)gw * FBAND + f];
    float  p = phi[f] * invLen;
    cf[jj] = make_float2(c.x * p, c.y * p);
    fv[jj] = (float)f;
  }
  float acc = 0.f;
  for (int k = 0; k < 16; ++k) {
    float s = 0.f;
#pragma unroll
    for (int jj = 0; jj < 3; ++jj) {
      float sn, cs; __sincosf((PI_F / 8.f) * fv[jj] * (float)k, &sn, &cs);
      s += cf[jj].x * cs - cf[jj].y * sn;
    }
    s = wave_all_sum(s);
    acc += logf(fabsf(s) + 1e-6f);
  }
  if (lane == 0) {
    int b = gw / perB, n = gw % perB;
    Fout[(size_t)b * NFEAT + featOff + n] = acc * (1.f / 16.f);
  }
}

// ---------------- stage-2 pair modulation ----------------
__global__ void k_mod2(const float2* __restrict__ U1sf, const float* __restrict__ psi2,
                       const int* __restrict__ pi1, const int* __restrict__ pi2,
                       float2* __restrict__ Z) {
  long long i = (long long)blockIdx.x * blockDim.x + threadIdx.x;
  if (i >= (long long)BNP * TT2) return;
  int f  = (int)(i % TT2);
  int bp = (int)(i / TT2);
  int b = bp / NPAIRS, p = bp % NPAIRS;
  float2 u = U1sf[((size_t)b * NN1 + pi1[p]) * TT2 + f];
  float  s = psi2[(size_t)pi2[p] * TT2 + f];
  Z[i] = make_float2(u.x * s, u.y * s);
}

__global__ void k_abs2(const float2* __restrict__ in, float* __restrict__ U2) {
  long long i = (long long)blockIdx.x * blockDim.x + threadIdx.x;
  if (i >= (long long)BNP * TT2) return;
  float2 v = in[i];
  U2[i] = sqrtf(v.x * v.x + v.y * v.y) * (1.f / (float)TT2);
}

// ---------------- zero fill ----------------
__global__ void k_zero(float* __restrict__ p, long long n) {
  long long i = (long long)blockIdx.x * blockDim.x + threadIdx.x;
  if (i < n) p[i] = 0.f;
}

// ---------------- pad weights: W (N x K) -> Wp (Np x Kp), zero elsewhere ----------------
__global__ void k_padw(const float* __restrict__ W, float* __restrict__ Wp,
                       int N, int K, int Np, int Kp) {
  long long i = (long long)blockIdx.x * blockDim.x + threadIdx.x;
  if (i >= (long long)Np * Kp) return;
  int k = (int)(i % Kp);
  int n = (int)(i / Kp);
  Wp[i] = (n < N && k < K) ? W[(size_t)n * K + k] : 0.f;
}

// ---------------- batchnorm over batch dim, writes into padded 32 x KP1 ----------------
__global__ void k_bn(const float* __restrict__ F, const float* __restrict__ gamma,
                     const float* __restrict__ beta, float* __restrict__ FBNp) {
  int feat = blockIdx.x;       // 673 blocks
  int lane = threadIdx.x;      // 32 lanes = batch index
  float v = F[(size_t)lane * NFEAT + feat];
  float mean = wave_all_sum(v) * (1.f / 32.f);
  float d = v - mean;
  float var = wave_all_sum(d * d) * (1.f / 32.f);
  float y = d * rsqrtf(var + 1e-5f) * gamma[feat] + beta[feat];
  FBNp[(size_t)lane * KP1 + feat] = y;
}

// ---------------- WMMA f32 GEMM on padded operands: Out = relu(A @ W^T + b) ----------------
// A: 32 x Kp row-major (zero-padded), W: Np x Kp row-major (zero-padded rows/cols).
// Out: 32 x Np, padded cols written as 0 so it can feed the next layer directly.
// One wave per 16x16 tile; Kp % 16 == 0 -> fully branch-free inner loop:
//   8x global_load_b64 + 4x v_wmma_f32_16x16x4_f32 per 16 K.
// Fragment layout (CDNA5 ISA): A 16x4 / B 4x16: K = vgpr + 2*(lane>=16), row/col = lane%16.
__global__ void k_gemm_wmma(const float* __restrict__ A, const float* __restrict__ W,
                            const float* __restrict__ bias, float* __restrict__ Out,
                            int Kp, int Np, int Nreal, int do_relu) {
  int tilesN = Np >> 4;
  int tm = blockIdx.x / tilesN, tn = blockIdx.x % tilesN;
  int lane = threadIdx.x;
  int half = lane >> 4;
  int l16  = lane & 15;
  const float* arow = A + (size_t)(tm * 16 + l16) * Kp + 2 * half;
  const float* brow = W + (size_t)(tn * 16 + l16) * Kp + 2 * half;
  v8f acc = {};
  for (int k0 = 0; k0 < Kp; k0 += 16) {
    v2f a0 = *(const v2f*)(arow + k0);
    v2f b0 = *(const v2f*)(brow + k0);
    v2f a1 = *(const v2f*)(arow + k0 + 4);
    v2f b1 = *(const v2f*)(brow + k0 + 4);
    v2f a2 = *(const v2f*)(arow + k0 + 8);
    v2f b2 = *(const v2f*)(brow + k0 + 8);
    v2f a3 = *(const v2f*)(arow + k0 + 12);
    v2f b3 = *(const v2f*)(brow + k0 + 12);
    acc = __builtin_amdgcn_wmma_f32_16x16x4_f32(false, a0, false, b0, (short)0, acc, false, false);
    acc = __builtin_amdgcn_wmma_f32_16x16x4_f32(false, a1, false, b1, (short)0, acc, false, false);
    acc = __builtin_amdgcn_wmma_f32_16x16x4_f32(false, a2, false, b2, (short)0, acc, false, false);
    acc = __builtin_amdgcn_wmma_f32_16x16x4_f32(false, a3, false, b3, (short)0, acc, false, false);
  }
#pragma unroll
  for (int v = 0; v < 8; ++v) {
    int m = tm * 16 + v + 8 * half;
    int n = tn * 16 + l16;
    float r = 0.f;
    if (n < Nreal) {
      r = acc[v] + bias[n];
      if (do_relu) r = fmaxf(r, 0.f);
    }
    Out[(size_t)m * Np + n] = r;   // padded cols -> 0
  }
}

// ---------------- layer 3 + log_softmax (35 logits / row) ----------------
__global__ void k_final(const float* __restrict__ H2, const float* __restrict__ W3,
                        const float* __restrict__ b3, float* __restrict__ out) {
  __shared__ float logits[35];
  __shared__ float red[2];
  int b = blockIdx.x, t = threadIdx.x;
  if (t < 35) {
    float s = b3[t];
    for (int k = 0; k < 90; ++k) s = fmaf(H2[(size_t)b * NP2 + k], W3[(size_t)t * 90 + k], s);
    logits[t] = s;
  }
  __syncthreads();
  if (t == 0) {
    float mx = logits[0];
    for (int i = 1; i < 35; ++i) mx = fmaxf(mx, logits[i]);
    float sum = 0.f;
    for (int i = 0; i < 35; ++i) sum += expf(logits[i] - mx);
    red[0] = mx; red[1] = logf(sum);
  }
  __syncthreads();
  if (t < 35) out[(size_t)b * 35 + t] = logits[t] - red[0] - red[1];
}

// ---------------- host-side FFT driver ----------------
static void run_fft(float2* a, float2* b, int N, int nbatch, float sign,
                    const int* rad, int np, hipStream_t s, float2** res) {
  float2* src = a; float2* dst = b; int Ns = 1;
  for (int i = 0; i < np; ++i) {
    int R = rad[i];
    long long total = (long long)nbatch * (N / R);
    int blocks = (int)((total + 255) / 256);
    k_stockham<<<blocks, 256, 0, s>>>(src, dst, N, Ns, R, nbatch, sign);
    Ns *= R;
    float2* tmp = src; src = dst; dst = tmp;
  }
  *res = src;
}

static inline int blks(long long n) { return (int)((n + 255) / 256); }

extern "C" void kernel_launch(void* const* d_in, const int* in_sizes, int n_in,
                              void* d_out, int out_size, void* d_ws, size_t ws_size,
                              hipStream_t stream) {
  (void)in_sizes; (void)n_in; (void)out_size;
  const float* x     = (const float*)d_in[0];
  const float* psi1  = (const float*)d_in[1];
  const float* phi1  = (const float*)d_in[2];
  const float* psi2  = (const float*)d_in[3];
  const float* phi2  = (const float*)d_in[4];
  const float* gamma = (const float*)d_in[5];
  const float* beta  = (const float*)d_in[6];
  const float* W1    = (const float*)d_in[7];
  const float* b1    = (const float*)d_in[8];
  const float* W2    = (const float*)d_in[9];
  const float* b2    = (const float*)d_in[10];
  const float* W3    = (const float*)d_in[11];
  const float* b3    = (const float*)d_in[12];
  const int*   pi1   = (const int*)d_in[13];
  const int*   pi2   = (const int*)d_in[14];
  float* out = (float*)d_out;

  // ---- workspace layout (512B aligned), total ~1.86 GiB ----
  char* base = (char*)d_ws;
  size_t off = 0;
  auto take = [&](size_t bytes) -> char* {
    char* p = base + off;
    off = (off + bytes + 511) & ~(size_t)511;
    return p;
  };
  float2* XF   = (float2*)take((size_t)BB * TT * sizeof(float2));     //   4.1 MB
  float2* BUFA = (float2*)take((size_t)BN1 * TT * sizeof(float2));    // 491.5 MB
  float2* BUFB = (float2*)take((size_t)BN1 * TT * sizeof(float2));    // 491.5 MB
  float*  U1   = (float*) take((size_t)BN1 * TT * sizeof(float));     // 245.8 MB
  float*  U1S  = (float*) take((size_t)BN1 * TT2 * sizeof(float));    //  30.7 MB
  float2* SA   = (float2*)take((size_t)BNP * TT2 * sizeof(float2));   // 283.1 MB
  float2* SB   = (float2*)take((size_t)BNP * TT2 * sizeof(float2));   // 283.1 MB
  float*  U2   = (float*) take((size_t)BNP * TT2 * sizeof(float));    // 141.6 MB
  float2* C1   = (float2*)take((size_t)BN1 * FBAND * sizeof(float2)); //   2.9 MB
  float2* C2   = (float2*)take((size_t)BNP * FBAND * sizeof(float2)); //  13.6 MB
  float*  Ff   = (float*) take((size_t)BB * NFEAT * sizeof(float));
  float*  FBNp = (float*) take((size_t)BB * KP1 * sizeof(float));     // 32 x 688
  float*  W1p  = (float*) take((size_t)NP1 * KP1 * sizeof(float));    // 304 x 688
  float*  W2p  = (float*) take((size_t)NP2 * NP1 * sizeof(float));    // 96 x 304
  float*  H1p  = (float*) take((size_t)BB * NP1 * sizeof(float));     // 32 x 304
  float*  H2p  = (float*) take((size_t)BB * NP2 * sizeof(float));     // 32 x 96
  if (off > ws_size) return; // workspace too small; nothing safe to do

  static const int rad16000[10] = {5, 5, 5, 2, 2, 2, 2, 2, 2, 2}; // 125*128
  static const int rad2000[7]   = {5, 5, 5, 2, 2, 2, 2};          // 125*16
  float2* res;

  // 1) Xf = fft(x): 32 x 16000 (10 passes, even -> result back in XF)
  k_r2c<<<blks((long long)BB * TT), 256, 0, stream>>>(x, XF, (long long)BB * TT);
  run_fft(XF, BUFA, TT, BB, -1.f, rad16000, 10, stream, &res); // res == XF

  // 2) U1 = |ifft(Xf * psi1)| : 3840 x 16000 (10 passes -> back in BUFA)
  k_mod1<<<blks((long long)BN1 * TT), 256, 0, stream>>>(res, psi1, BUFA);
  run_fft(BUFA, BUFB, TT, BN1, +1.f, rad16000, 10, stream, &res); // res == BUFA
  k_abs1<<<blks((long long)BN1 * TT), 256, 0, stream>>>(res, U1, U1S);

  // 3) S1 features
  k_banddft<<<blks((long long)BN1 * FBAND * 32), 256, 0, stream>>>(U1, C1, TT, BN1);
  k_finalize<<<blks((long long)BN1 * 32), 256, 0, stream>>>(
      C1, phi1, 1.f / (float)TT, BN1, NN1, 0, Ff);

  // 4) U1sf = fft(U1[::8]) : 3840 x 2000 (7 passes, odd -> result in SB head)
  k_r2c<<<blks((long long)BN1 * TT2), 256, 0, stream>>>(U1S, SA, (long long)BN1 * TT2);
  run_fft(SA, SB, TT2, BN1, -1.f, rad2000, 7, stream, &res); // res == SB (head)

  // 5) U2 = |ifft(U1sf[pairs] * psi2[pairs])| : 17696 x 2000 (7 passes -> SB)
  k_mod2<<<blks((long long)BNP * TT2), 256, 0, stream>>>(res, psi2, pi1, pi2, SA);
  run_fft(SA, SB, TT2, BNP, +1.f, rad2000, 7, stream, &res); // res == SB
  k_abs2<<<blks((long long)BNP * TT2), 256, 0, stream>>>(res, U2);

  // 6) S2 features
  k_banddft<<<blks((long long)BNP * FBAND * 32), 256, 0, stream>>>(U2, C2, TT2, BNP);
  k_finalize<<<blks((long long)BNP * 32), 256, 0, stream>>>(
      C2, phi2, 1.f / (float)TT2, BNP, NPAIRS, 120, Ff);

  // 7) pad weights + activations (branch-free WMMA operands)
  k_padw<<<blks((long long)NP1 * KP1), 256, 0, stream>>>(W1, W1p, 300, NFEAT, NP1, KP1);
  k_padw<<<blks((long long)NP2 * NP1), 256, 0, stream>>>(W2, W2p, 90, 300, NP2, NP1);
  k_zero<<<blks((long long)BB * KP1), 256, 0, stream>>>(FBNp, (long long)BB * KP1);
  k_bn<<<NFEAT, 32, 0, stream>>>(Ff, gamma, beta, FBNp);

  // 8) WMMA MLP + log_softmax
  k_gemm_wmma<<<2 * (NP1 / 16), 32, 0, stream>>>(FBNp, W1p, b1, H1p, KP1, NP1, 300, 1);
  k_gemm_wmma<<<2 * (NP2 / 16), 32, 0, stream>>>(H1p, W2p, b2, H2p, NP1, NP2, 90, 1);
  k_final<<<BB, 64, 0, stream>>>(H2p, W3, b3, out);
}